// NeuralNet_11252814315873
// MI455X (gfx1250) — compile-verified
//
#include <hip/hip_runtime.h>
#include <math.h>

typedef float v2f __attribute__((ext_vector_type(2)));
typedef float v8f __attribute__((ext_vector_type(8)));

#define NN      512
#define NCAND   7
#define DM      300
#define KREL    3
#define MFLAT   (NN*NCAND)      /* 3584 */
#define ER_LD   304
#define ER_KSZ  (MFLAT*ER_LD)   /* 1089536 per k */
#define RT_KSZ  (ER_LD*DM)      /* 91200 per k: RT[k][e(304)][d(300)] */
#define DAMP    0.5f

/* ---------------- workspace layout (float offsets) ---------------- */
#define OFF_BT   0u                       /* 300*300            =  90000 */
#define OFF_DT   90000u                   /* 3*300*300          = 270000 */
#define OFF_PSI  360000u                  /* 512*8              =   4096 */
#define OFF_FK   364096u                  /* 3*300*512 (FkT)    = 460800 */
#define OFF_AW   824896u                  /* 512*512*4          = 1048576 */
#define OFF_ER   1873472u                 /* 3*3584*304         = 3268608 */
#define OFF_L    5142080u                 /* 512*512*49         = 12845056 */
#define OFF_M0   17987136u                /* 512*512*8          = 2097152 */
#define OFF_M1   20084288u                /* 512*512*8          = 2097152 */
#define OFF_T    22181440u                /* 512*8              =   4096 */
#define OFF_RT   22185536u                /* 3*304*300          = 273600 */
/* total 22459136 floats ~ 89.8 MB */

/* ---- transpose B (300x300) and D (3x300x300) for coalesced GEMV ---- */
__global__ void k_transpose(const float* __restrict__ B, const float* __restrict__ D,
                            float* __restrict__ Bt, float* __restrict__ Dt) {
    int id = blockIdx.x * 256 + threadIdx.x;
    if (id >= 4 * 90000) return;
    int mat = id / 90000, r = id - mat * 90000;
    int d = r / 300, e = r - d * 300;
    float v = (mat == 0) ? B[d * 300 + e] : D[(mat - 1) * 90000 + d * 300 + e];
    if (mat == 0) Bt[e * 300 + d] = v;
    else          Dt[(mat - 1) * 90000 + e * 300 + d] = v;
}

/* ---- RT[k][e][d] = R[k][d][e], e zero-padded to 304 (for guard-free WMMA B-frags) ---- */
__global__ void k_padRT(const float* __restrict__ R, float* __restrict__ RT) {
    int id = blockIdx.x * 256 + threadIdx.x;
    if (id >= KREL * RT_KSZ) return;
    int k = id / RT_KSZ, r = id - k * RT_KSZ;
    int e = r / DM, d = r - e * DM;
    RT[id] = (e < DM) ? R[k * 90000 + d * DM + e] : 0.f;
}

/* ---- psi[n,c] = emb[n,c,:] . (B fmc[n]) ---- */
__global__ void k_psi(const float* __restrict__ emb, const float* __restrict__ fmcs,
                      const float* __restrict__ Bt, float* __restrict__ psi) {
    int n = blockIdx.x, tid = threadIdx.x;
    __shared__ float fm[DM];
    __shared__ float g[DM];
    if (tid < DM) fm[tid] = fmcs[n * DM + tid];
    __syncthreads();
    if (tid < DM) {
        float s = 0.f;
        for (int e = 0; e < DM; ++e) s += Bt[e * 300 + tid] * fm[e];   /* coalesced over tid */
        g[tid] = s;
    }
    __syncthreads();
    int c = tid >> 5, lane = tid & 31;
    if (c < NCAND) {
        float s = 0.f;
        for (int d = lane; d < DM; d += 32) s += emb[(n * NCAND + c) * DM + d] * g[d];
        #pragma unroll
        for (int off = 16; off > 0; off >>= 1) s += __shfl_xor(s, off, 32);
        if (lane == 0) psi[n * 8 + c] = s;
    }
}

/* ---- FkT[k][d][j] = sum_e D[k][d][e] fmc[j][e] (transposed output) ---- */
__global__ void k_fk(const float* __restrict__ fmcs, const float* __restrict__ Dt,
                     float* __restrict__ FkT) {
    int j = blockIdx.x, k = blockIdx.y, tid = threadIdx.x;
    __shared__ float fm[DM];
    if (tid < DM) fm[tid] = fmcs[j * DM + tid];
    __syncthreads();
    if (tid < DM) {
        const float* Dk = Dt + k * 90000;
        float s = 0.f;
        for (int e = 0; e < DM; ++e) s += Dk[e * 300 + tid] * fm[e];
        FkT[(k * DM + tid) * NN + j] = s;
    }
}

/* ---- a[i,j,k] = softmax_k( fmc_i . FkT[k][:,j] / sqrt(d) ) ---- */
__global__ void k_eb(const float* __restrict__ fmcs, const float* __restrict__ FkT,
                     float* __restrict__ aw) {
    int i = blockIdx.x, tid = threadIdx.x;
    int j = blockIdx.y * 256 + tid;
    __shared__ float fm[DM];
    for (int e = tid; e < DM; e += 256) fm[e] = fmcs[i * DM + e];
    __syncthreads();
    float a0 = 0.f, a1 = 0.f, a2 = 0.f;
    for (int d = 0; d < DM; ++d) {
        float f = fm[d];
        int base = d * NN + j;
        a0 += f * FkT[base];
        a1 += f * FkT[DM * NN + base];
        a2 += f * FkT[2 * DM * NN + base];
    }
    const float inv_sqrt_d = 0.057735026918962584f;   /* 1/sqrt(300) */
    float e0 = expf(a0 * inv_sqrt_d), e1 = expf(a1 * inv_sqrt_d), e2 = expf(a2 * inv_sqrt_d);
    float inv = 1.f / (e0 + e1 + e2);
    float* p = aw + (size_t)(i * NN + j) * 4;
    p[0] = e0 * inv; p[1] = e1 * inv; p[2] = e2 * inv; p[3] = 0.f;
}

/* ---- WMMA GEMM: ER[k][m][e] = sum_d Emb[m][d] * RT[k][e][d]  (fp32 16x16x4) ---- */
__global__ void k_er(const float* __restrict__ emb, const float* __restrict__ RT,
                     float* __restrict__ ER) {
    int lane = threadIdx.x;
    int m0 = blockIdx.x * 16, e0c = blockIdx.y * 16, k = blockIdx.z;
    int half = lane >> 4, l15 = lane & 15;
    int row = m0 + l15, col = e0c + l15;                 /* col < 304 always valid in RT */
    const float* Arow = emb + (size_t)row * DM;
    const float* Bcol = RT + (size_t)k * RT_KSZ + (size_t)col * DM;
    v8f acc = {0.f,0.f,0.f,0.f,0.f,0.f,0.f,0.f};
    for (int d0 = 0; d0 < DM; d0 += 4) {
        int kk = d0 + (half << 1);
        v2f av = *(const v2f*)(Arow + kk);               /* A: 16x4 frag   */
        v2f bv = *(const v2f*)(Bcol + kk);               /* B: 4x16 frag   */
        acc = __builtin_amdgcn_wmma_f32_16x16x4_f32(false, av, false, bv,
                                                    (short)0, acc, false, false);
    }
    if (col < DM) {
        float* ERk = ER + k * ER_KSZ;
        #pragma unroll
        for (int v = 0; v < 8; ++v) {
            int m = m0 + v + (half << 3);
            ERk[m * ER_LD + col] = acc[v];
        }
    }
}

/* ---- WMMA GEMM (dominant): lbp[i][j][a][b] = sum_k a_w[i,j,k]*(ER_k[ia,:].emb[jb,:]) + psi[i,a] ---- */
__global__ void k_phi(const float* __restrict__ emb, const float* __restrict__ ER,
                      const float* __restrict__ aw, const float* __restrict__ psi,
                      float* __restrict__ L) {
    int lane = threadIdx.x;
    int m0 = blockIdx.x * 16, n0 = blockIdx.y * 16;
    int half = lane >> 4, l15 = lane & 15;
    const float* Brow = emb + (size_t)(n0 + l15) * DM;
    const float* A0 = ER + (size_t)(m0 + l15) * ER_LD;
    const float* A1 = A0 + ER_KSZ;
    const float* A2 = A1 + ER_KSZ;
    v8f acc0 = {0.f,0.f,0.f,0.f,0.f,0.f,0.f,0.f};
    v8f acc1 = acc0, acc2 = acc0;
    for (int e0 = 0; e0 < DM; e0 += 4) {
        int kk = e0 + (half << 1);
        v2f bv = *(const v2f*)(Brow + kk);
        v2f a0 = *(const v2f*)(A0 + kk);
        v2f a1 = *(const v2f*)(A1 + kk);
        v2f a2 = *(const v2f*)(A2 + kk);
        acc0 = __builtin_amdgcn_wmma_f32_16x16x4_f32(false, a0, false, bv, (short)0, acc0, false, false);
        acc1 = __builtin_amdgcn_wmma_f32_16x16x4_f32(false, a1, false, bv, (short)0, acc1, false, false);
        acc2 = __builtin_amdgcn_wmma_f32_16x16x4_f32(false, a2, false, bv, (short)0, acc2, false, false);
    }
    int n = n0 + l15;
    int j = n / 7, b = n - j * 7;
    #pragma unroll
    for (int v = 0; v < 8; ++v) {
        int m = m0 + v + (half << 3);
        int i = m / 7, a = m - i * 7;
        const float* p = aw + (size_t)(i * NN + j) * 4;
        float val = acc0[v] * p[0] + acc1[v] * p[1] + acc2[v] * p[2] + psi[i * 8 + a];
        L[((size_t)(i * NN + j) * 7 + a) * 7 + b] = val;
    }
}

/* ---- zero mbar buffer ---- */
__global__ void k_zero(float* __restrict__ buf) {
    int id = blockIdx.x * 256 + threadIdx.x;
    if (id < NN * NN * 8) buf[id] = 0.f;
}

/* ---- T[x][c] = sum_i mbar[i][x][c] ---- */
__global__ void k_colsum(const float* __restrict__ cur, float* __restrict__ T) {
    int x = blockIdx.x, tid = threadIdx.x;
    int tc = tid & 7, ti = tid >> 3;
    float s = 0.f;
    for (int i = ti; i < NN; i += 32) s += cur[(size_t)(i * NN + x) * 8 + tc];
    __shared__ float red[256];
    red[tid] = s;
    __syncthreads();
    for (int st = 128; st >= 8; st >>= 1) {
        if (tid < st) red[tid] += red[tid + st];
        __syncthreads();
    }
    if (tid < 8) T[x * 8 + tid] = red[tid];
}

/* ---- one LBP damped message update ---- */
__global__ void k_lbp_step(const float* __restrict__ cur, float* __restrict__ nxt,
                           const float* __restrict__ T, const float* __restrict__ L,
                           const int* __restrict__ lengths) {
    int id = blockIdx.x * 256 + threadIdx.x;
    if (id >= NN * NN) return;
    int i = id >> 9, j = id & 511;
    const float* Lb = L + (size_t)id * 49;
    float ta[7];
    #pragma unroll
    for (int a = 0; a < 7; ++a) ta[a] = T[i * 8 + a] - cur[(size_t)(j * NN + i) * 8 + a];
    float mv[7];
    #pragma unroll
    for (int b = 0; b < 7; ++b) {
        float m = Lb[b] + ta[0];
        #pragma unroll
        for (int a = 1; a < 7; ++a) m = fmaxf(m, Lb[a * 7 + b] + ta[a]);
        mv[b] = m;
    }
    int ncv = lengths[j] + 1;
    float mean = 0.f;
    #pragma unroll
    for (int b = 0; b < 7; ++b) if (b < ncv) mean += mv[b];
    mean /= (float)ncv;
    float ex[7]; float den = 0.f;
    #pragma unroll
    for (int b = 0; b < 7; ++b) { ex[b] = (b < ncv) ? expf(mv[b] - mean) : 0.f; den += ex[b]; }
    float inv = 1.f / den;
    const float* cb = cur + (size_t)id * 8;
    float* nb = nxt + (size_t)id * 8;
    #pragma unroll
    for (int c = 0; c < 7; ++c) {
        float nm = (c < ncv) ? (expf(cb[c]) * (1.f - DAMP) + DAMP * ex[c] * inv) : 1.f;
        nb[c] = logf(nm);
    }
    nb[7] = 0.f;
}

/* ---- final ubar ---- */
__global__ void k_final(const float* __restrict__ cur, const float* __restrict__ psi,
                        const int* __restrict__ lengths, float* __restrict__ out) {
    int x = blockIdx.x, tid = threadIdx.x;
    int tc = tid & 7, ti = tid >> 3;
    float s = 0.f;
    for (int i = ti; i < NN; i += 32)
        if (i != x) s += cur[(size_t)(i * NN + x) * 8 + tc];
    __shared__ float red[256];
    red[tid] = s;
    __syncthreads();
    for (int st = 128; st >= 8; st >>= 1) {
        if (tid < st) red[tid] += red[tid + st];
        __syncthreads();
    }
    if (tid == 0) {
        int ncv = lengths[x] + 1;
        float u[7];
        for (int c = 0; c < 7; ++c) u[c] = psi[x * 8 + c] + red[c];
        float mean = 0.f;
        for (int c = 0; c < 7; ++c) if (c < ncv) mean += u[c];
        mean /= (float)ncv;
        float ub[7]; float ssum = 0.f;
        for (int c = 0; c < 7; ++c) { ub[c] = (c < ncv) ? expf(u[c] - mean) : 0.f; ssum += ub[c]; }
        if (ssum == 0.f) ssum = 1.f;
        float inv = 1.f / ssum;
        for (int c = 0; c < 7; ++c) out[x * 7 + c] = ub[c] * inv;
    }
}

extern "C" void kernel_launch(void* const* d_in, const int* in_sizes, int n_in,
                              void* d_out, int out_size, void* d_ws, size_t ws_size,
                              hipStream_t stream) {
    (void)in_sizes; (void)n_in; (void)out_size; (void)ws_size;
    const float* emb    = (const float*)d_in[0];
    const float* fmcs   = (const float*)d_in[1];
    const float* B      = (const float*)d_in[2];
    const float* R      = (const float*)d_in[3];
    const float* Dm     = (const float*)d_in[4];
    const int*   lens   = (const int*)d_in[5];
    float* out = (float*)d_out;
    float* ws  = (float*)d_ws;

    float* Bt  = ws + OFF_BT;
    float* Dt  = ws + OFF_DT;
    float* psi = ws + OFF_PSI;
    float* FkT = ws + OFF_FK;
    float* aw  = ws + OFF_AW;
    float* ER  = ws + OFF_ER;
    float* L   = ws + OFF_L;
    float* m0  = ws + OFF_M0;
    float* m1  = ws + OFF_M1;
    float* T   = ws + OFF_T;
    float* RT  = ws + OFF_RT;

    k_transpose<<<dim3((4 * 90000 + 255) / 256), dim3(256), 0, stream>>>(B, Dm, Bt, Dt);
    k_padRT<<<dim3((KREL * RT_KSZ + 255) / 256), dim3(256), 0, stream>>>(R, RT);
    k_psi<<<dim3(NN), dim3(320), 0, stream>>>(emb, fmcs, Bt, psi);
    k_fk<<<dim3(NN, KREL), dim3(320), 0, stream>>>(fmcs, Dt, FkT);
    k_eb<<<dim3(NN, 2), dim3(256), 0, stream>>>(fmcs, FkT, aw);
    k_er<<<dim3(MFLAT / 16, ER_LD / 16, KREL), dim3(32), 0, stream>>>(emb, RT, ER);
    k_phi<<<dim3(MFLAT / 16, MFLAT / 16), dim3(32), 0, stream>>>(emb, ER, aw, psi, L);
    k_zero<<<dim3((NN * NN * 8) / 256), dim3(256), 0, stream>>>(m0);

    float* cur = m0;
    float* nxt = m1;
    for (int it = 0; it < 10; ++it) {
        k_colsum<<<dim3(NN), dim3(256), 0, stream>>>(cur, T);
        k_lbp_step<<<dim3((NN * NN) / 256), dim3(256), 0, stream>>>(cur, nxt, T, L, lens);
        float* tmp = cur; cur = nxt; nxt = tmp;
    }
    k_final<<<dim3(NN), dim3(256), 0, stream>>>(cur, psi, lens, out);
}